// Net_75230647156964
// MI455X (gfx1250) — compile-verified
//
#include <hip/hip_runtime.h>

// CDNA5 (gfx1250) fused MoE forward, all matmuls via V_WMMA_F32_16X16X4_F32
// (exact f32 — matches the f32 reference precision path).
//
// Layouts (per ISA 7.12.2, wave32):
//   A 16x4 f32 : lane -> M = lane&15 ; VGPR v -> K = v + 2*(lane>=16)
//   B 4x16 f32 : lane -> N = lane&15 ; VGPR v -> K = v + 2*(lane>=16)  (mirror of A)
//   D 16x16 f32: lane -> N = lane&15 ; VGPR r -> M = r + 8*(lane>=16)
// We put weights in A (M = feature) and activations in B (N = row), so D holds
// rows on lanes -> per-row epilogue after an LDS roundtrip.

typedef __attribute__((ext_vector_type(2))) float v2f;
typedef __attribute__((ext_vector_type(8))) float v8f;

#define LN_EPS 1e-5f

__device__ __forceinline__ float silu_f(float v) { return v / (1.0f + __expf(-v)); }

__device__ __forceinline__ v8f wmma4(v2f a, v2f b, v8f c) {
  // (neg_a, A, neg_b, B, c_mod, C, reuse_a, reuse_b)
  return __builtin_amdgcn_wmma_f32_16x16x4_f32(false, a, false, b, (short)0, c, false, false);
}

__global__ __launch_bounds__(128) void moe_fused_wmma(
    const float* __restrict__ x,
    const float* __restrict__ W1, const float* __restrict__ b1,
    const float* __restrict__ ln_g, const float* __restrict__ ln_b,
    const float* __restrict__ W2, const float* __restrict__ b2,
    const float* __restrict__ Wg1, const float* __restrict__ bg1,
    const float* __restrict__ Wg2, const float* __restrict__ bg2,
    const float* __restrict__ We1, const float* __restrict__ be1,
    const float* __restrict__ We2, const float* __restrict__ be2,
    const float* __restrict__ Wh, const float* __restrict__ bh,
    const float* __restrict__ outscale,
    float* __restrict__ out, int nrows)
{
  // per-wave activation/transpose tile: 32 rows x 68 (stride 68 => bank-safe)
  __shared__ float sBuf[4 * 32 * 68];                       // 34816 B
  __shared__ float sWcatT[48 * 68];  // [feature][k] K=64    13056 B
  __shared__ float sW2T[32 * 28];    // [feature][k] K=24     3584 B
  __shared__ float sWe1T[64 * 28];   // [feature][k] K=24     7168 B
  __shared__ float sWe2T[16 * 68];   // [feature][k] K=64     4352 B
  __shared__ float sB1[24], sLnG[24], sLnB[24], sB2[24];
  __shared__ float sBg1[16], sWg2[48], sBg2[3];
  __shared__ float sBe1[60], sBe2[9], sBh[3], sOutS[3];

  const int tid = threadIdx.x;

  // ---- cooperative weight staging (transposed so A-operand is a ds_load_b64) ----
  for (int idx = tid; idx < 48 * 64; idx += 128) {          // [Wcat]^T = [W1|Wg1|Wh|0]
    int m = idx >> 6, k = idx & 63;
    float v;
    if (m < 24)      v = W1[k * 24 + m];
    else if (m < 40) v = Wg1[k * 16 + (m - 24)];
    else if (m < 43) v = Wh[k * 3 + (m - 40)];
    else             v = 0.0f;
    sWcatT[m * 68 + k] = v;
  }
  for (int idx = tid; idx < 32 * 24; idx += 128) {          // W2^T (out padded to 32)
    int j = idx / 24, k = idx - j * 24;
    sW2T[j * 28 + k] = (j < 24) ? W2[k * 24 + j] : 0.0f;
  }
  for (int idx = tid; idx < 64 * 24; idx += 128) {          // [We1 concat]^T (60->64)
    int f = idx / 24, k = idx - f * 24;
    float v = 0.0f;
    if (f < 60) v = We1[(f / 20) * 480 + k * 20 + (f % 20)];
    sWe1T[f * 28 + k] = v;
  }
  for (int idx = tid; idx < 16 * 64; idx += 128) {          // block-diag We2^T (60x9 -> 64x16)
    int f = idx >> 6, k = idx & 63;
    float v = 0.0f;
    if (f < 9 && k < 60) {
      int e = k / 20;
      if (e == f / 3) v = We2[e * 60 + (k % 20) * 3 + (f % 3)];
    }
    sWe2T[f * 68 + k] = v;
  }
  if (tid < 24) { sB1[tid] = b1[tid]; sLnG[tid] = ln_g[tid]; sLnB[tid] = ln_b[tid]; sB2[tid] = b2[tid]; }
  if (tid < 16) sBg1[tid] = bg1[tid];
  if (tid < 48) sWg2[tid] = Wg2[tid];
  if (tid < 60) sBe1[tid] = be1[tid];
  if (tid < 9)  sBe2[tid] = be2[tid];
  if (tid < 3)  { sBg2[tid] = bg2[tid]; sBh[tid] = bh[tid]; sOutS[tid] = outscale[tid]; }

  const int wave = tid >> 5;
  const int lane = tid & 31;
  const int lo = lane & 15;
  const int hi = lane >> 4;
  float* buf = sBuf + wave * (32 * 68);
  const long rowbase = (long)blockIdx.x * 128 + (long)wave * 32;

  // ---- stage this wave's 32x64 x-tile into LDS (coalesced float4) ----
  for (int j = 0; j < 16; ++j) {
    int linear = j * 32 + lane;
    int r = linear >> 4, c4 = linear & 15;
    long grow = rowbase + r;
    if (grow > (long)nrows - 1) grow = (long)nrows - 1;
    const float4 v = *(const float4*)(x + grow * 64 + c4 * 4);
    *(float4*)(buf + r * 68 + c4 * 4) = v;
  }
  __syncthreads();

  // ================= Stage 1: X[32x64] @ Wcat[64x48] =================
  v8f acc1[2][3] = {};
  #pragma unroll
  for (int s = 0; s < 16; ++s) {
    const int kk = 4 * s + 2 * hi;
    v2f a0 = *(const v2f*)&sWcatT[(0 * 16 + lo) * 68 + kk];
    v2f a1 = *(const v2f*)&sWcatT[(1 * 16 + lo) * 68 + kk];
    v2f a2 = *(const v2f*)&sWcatT[(2 * 16 + lo) * 68 + kk];
    v2f bb0 = *(const v2f*)&buf[(0 * 16 + lo) * 68 + kk];
    v2f bb1 = *(const v2f*)&buf[(1 * 16 + lo) * 68 + kk];
    acc1[0][0] = wmma4(a0, bb0, acc1[0][0]);
    acc1[0][1] = wmma4(a1, bb0, acc1[0][1]);
    acc1[0][2] = wmma4(a2, bb0, acc1[0][2]);
    acc1[1][0] = wmma4(a0, bb1, acc1[1][0]);
    acc1[1][1] = wmma4(a1, bb1, acc1[1][1]);
    acc1[1][2] = wmma4(a2, bb1, acc1[1][2]);
  }
  // D -> LDS, per-row layout (cols 0..47 overwrite the consumed x-tile)
  #pragma unroll
  for (int t = 0; t < 2; ++t) {
    const int rb = (t * 16 + lo) * 68;
    #pragma unroll
    for (int n = 0; n < 3; ++n)
      #pragma unroll
      for (int r = 0; r < 8; ++r)
        buf[rb + 16 * n + r + 8 * hi] = acc1[t][n][r];
  }
  __syncthreads();

  // ====== per-lane roundtrip: SiLU + LayerNorm, gate, head (lane = row) ======
  float g0, g1, g2, hd0, hd1, hd2;
  {
    const int rb = lane * 68;
    float h1v[24];
    float ssum = 0.0f;
    #pragma unroll
    for (int j = 0; j < 24; ++j) {
      float p = buf[rb + j] + sB1[j];
      float sv = silu_f(p);
      h1v[j] = sv; ssum += sv;
    }
    float mu = ssum * (1.0f / 24.0f);
    float var = 0.0f;
    #pragma unroll
    for (int j = 0; j < 24; ++j) { float d = h1v[j] - mu; var += d * d; }
    float rn = rsqrtf(var * (1.0f / 24.0f) + LN_EPS);
    #pragma unroll
    for (int j = 0; j < 24; ++j)
      buf[rb + j] = (h1v[j] - mu) * rn * sLnG[j] + sLnB[j];
    // gate: tanh(x@Wg1+bg1) @ Wg2 + bg2 -> softmax
    float gh[16];
    #pragma unroll
    for (int i = 0; i < 16; ++i) gh[i] = tanhf(buf[rb + 24 + i] + sBg1[i]);
    float l0 = sBg2[0], l1 = sBg2[1], l2 = sBg2[2];
    #pragma unroll
    for (int i = 0; i < 16; ++i) {
      l0 += gh[i] * sWg2[i * 3 + 0];
      l1 += gh[i] * sWg2[i * 3 + 1];
      l2 += gh[i] * sWg2[i * 3 + 2];
    }
    float mx = fmaxf(l0, fmaxf(l1, l2));
    float e0 = __expf(l0 - mx), e1 = __expf(l1 - mx), e2 = __expf(l2 - mx);
    float inv = 1.0f / (e0 + e1 + e2);
    g0 = e0 * inv; g1 = e1 * inv; g2 = e2 * inv;
    hd0 = buf[rb + 40] + sBh[0];
    hd1 = buf[rb + 41] + sBh[1];
    hd2 = buf[rb + 42] + sBh[2];
  }
  __syncthreads();

  // ================= Stage 2: hn[32x24] @ W2[24x24] =================
  v8f acc2[2][2] = {};
  #pragma unroll
  for (int s = 0; s < 6; ++s) {
    const int kk = 4 * s + 2 * hi;
    v2f a0 = *(const v2f*)&sW2T[(0 * 16 + lo) * 28 + kk];
    v2f a1 = *(const v2f*)&sW2T[(1 * 16 + lo) * 28 + kk];
    v2f bb0 = *(const v2f*)&buf[(0 * 16 + lo) * 68 + kk];
    v2f bb1 = *(const v2f*)&buf[(1 * 16 + lo) * 68 + kk];
    acc2[0][0] = wmma4(a0, bb0, acc2[0][0]);
    acc2[0][1] = wmma4(a1, bb0, acc2[0][1]);
    acc2[1][0] = wmma4(a0, bb1, acc2[1][0]);
    acc2[1][1] = wmma4(a1, bb1, acc2[1][1]);
  }
  #pragma unroll
  for (int t = 0; t < 2; ++t) {
    const int rb = (t * 16 + lo) * 68;
    #pragma unroll
    for (int n = 0; n < 2; ++n)
      #pragma unroll
      for (int r = 0; r < 8; ++r) {
        int feat = 16 * n + r + 8 * hi;
        if (feat < 24) {
          float p = acc2[t][n][r] + sB2[feat];
          buf[rb + feat] = silu_f(p);
        }
      }
  }
  __syncthreads();

  // ====== Stage 3: h2[32x24] @ We1cat[24x60] (+be1, SiLU elementwise) ======
  v8f acc3[2][4] = {};
  #pragma unroll
  for (int s = 0; s < 6; ++s) {
    const int kk = 4 * s + 2 * hi;
    v2f a0 = *(const v2f*)&sWe1T[(0 * 16 + lo) * 28 + kk];
    v2f a1 = *(const v2f*)&sWe1T[(1 * 16 + lo) * 28 + kk];
    v2f a2 = *(const v2f*)&sWe1T[(2 * 16 + lo) * 28 + kk];
    v2f a3 = *(const v2f*)&sWe1T[(3 * 16 + lo) * 28 + kk];
    v2f bb0 = *(const v2f*)&buf[(0 * 16 + lo) * 68 + kk];
    v2f bb1 = *(const v2f*)&buf[(1 * 16 + lo) * 68 + kk];
    acc3[0][0] = wmma4(a0, bb0, acc3[0][0]);
    acc3[0][1] = wmma4(a1, bb0, acc3[0][1]);
    acc3[0][2] = wmma4(a2, bb0, acc3[0][2]);
    acc3[0][3] = wmma4(a3, bb0, acc3[0][3]);
    acc3[1][0] = wmma4(a0, bb1, acc3[1][0]);
    acc3[1][1] = wmma4(a1, bb1, acc3[1][1]);
    acc3[1][2] = wmma4(a2, bb1, acc3[1][2]);
    acc3[1][3] = wmma4(a3, bb1, acc3[1][3]);
  }
  #pragma unroll
  for (int t = 0; t < 2; ++t) {
    const int rb = (t * 16 + lo) * 68;
    #pragma unroll
    for (int n = 0; n < 4; ++n)
      #pragma unroll
      for (int r = 0; r < 8; ++r) {
        int feat = 16 * n + r + 8 * hi;
        float v = 0.0f;                    // cols 60..63 become K-padding zeros
        if (feat < 60) v = silu_f(acc3[t][n][r] + sBe1[feat]);
        buf[rb + feat] = v;
      }
  }
  __syncthreads();

  // ====== Stage 4: eh[32x64(pad)] @ We2_blockdiag[64x9(pad16)] ======
  v8f acc4[2] = {};
  #pragma unroll
  for (int s = 0; s < 16; ++s) {
    const int kk = 4 * s + 2 * hi;
    v2f a0 = *(const v2f*)&sWe2T[lo * 68 + kk];
    v2f bb0 = *(const v2f*)&buf[(0 * 16 + lo) * 68 + kk];
    v2f bb1 = *(const v2f*)&buf[(1 * 16 + lo) * 68 + kk];
    acc4[0] = wmma4(a0, bb0, acc4[0]);
    acc4[1] = wmma4(a0, bb1, acc4[1]);
  }
  #pragma unroll
  for (int t = 0; t < 2; ++t) {
    const int rb = (t * 16 + lo) * 68;
    #pragma unroll
    for (int r = 0; r < 8; ++r) {
      int feat = r + 8 * hi;
      if (feat < 9) buf[rb + feat] = acc4[t][r];
    }
  }
  __syncthreads();

  // ====== final per-lane combine: out = head + (gate . e_out) * out_scale ======
  {
    const int rb = lane * 68;
    float eo[9];
    #pragma unroll
    for (int f = 0; f < 9; ++f) eo[f] = buf[rb + f] + sBe2[f];
    float m0 = g0 * eo[0] + g1 * eo[3] + g2 * eo[6];
    float m1 = g0 * eo[1] + g1 * eo[4] + g2 * eo[7];
    float m2 = g0 * eo[2] + g1 * eo[5] + g2 * eo[8];
    long grow = rowbase + lane;
    if (grow < (long)nrows) {
      out[grow * 3 + 0] = hd0 + m0 * sOutS[0];
      out[grow * 3 + 1] = hd1 + m1 * sOutS[1];
      out[grow * 3 + 2] = hd2 + m2 * sOutS[2];
    }
  }
}

extern "C" void kernel_launch(void* const* d_in, const int* in_sizes, int n_in,
                              void* d_out, int out_size, void* d_ws, size_t ws_size,
                              hipStream_t stream) {
  const float* x    = (const float*)d_in[0];
  const float* W1   = (const float*)d_in[1];
  const float* b1   = (const float*)d_in[2];
  const float* ln_g = (const float*)d_in[3];
  const float* ln_b = (const float*)d_in[4];
  const float* W2   = (const float*)d_in[5];
  const float* b2   = (const float*)d_in[6];
  const float* Wg1  = (const float*)d_in[7];
  const float* bg1  = (const float*)d_in[8];
  const float* Wg2  = (const float*)d_in[9];
  const float* bg2  = (const float*)d_in[10];
  const float* We1  = (const float*)d_in[11];
  const float* be1  = (const float*)d_in[12];
  const float* We2  = (const float*)d_in[13];
  const float* be2  = (const float*)d_in[14];
  const float* Wh   = (const float*)d_in[15];
  const float* bh   = (const float*)d_in[16];
  const float* os   = (const float*)d_in[17];
  const int nrows = in_sizes[0] / 64;

  dim3 grid((nrows + 127) / 128), block(128);
  hipLaunchKernelGGL(moe_fused_wmma, grid, block, 0, stream,
                     x, W1, b1, ln_g, ln_b, W2, b2, Wg1, bg1, Wg2, bg2,
                     We1, be1, We2, be2, Wh, bh, os, (float*)d_out, nrows);
}